// GraphSAGE_27376121545090
// MI455X (gfx1250) — compile-verified
//
#include <hip/hip_runtime.h>

// ---------------------------------------------------------------------------
// GraphSAGE (2-layer, mean aggregator) for MI455X / gfx1250.
//   N=50000 nodes, E=1.6M edges, 128 features.
//   Sparse aggregate: wave-per-edge float4 gather + native fp32 atomic scatter
//     (agent-scope relaxed -> global_atomic_add_f32; all traffic L2-resident:
//      feature table is 25.6 MB vs 192 MB L2).
//   Dense part: fp32 WMMA (v_wmma_f32_16x16x4_f32), 16x16 tile per wave,
//     keeps full fp32 reference precision.
// ---------------------------------------------------------------------------

typedef float v2f __attribute__((ext_vector_type(2)));
typedef float v8f __attribute__((ext_vector_type(8)));

#define NNODES 50000
#define FEATS  128

// Native device-scope fp32 atomic add: relaxed + agent scope lowers to
// global_atomic_add_f32 (no CAS loop, no fences).
__device__ __forceinline__ void atomic_add_f32(float* p, float v) {
  __hip_atomic_fetch_add(p, v, __ATOMIC_RELAXED, __HIP_MEMORY_SCOPE_AGENT);
}

// ---------------------------------------------------------------- degree ---
__global__ void deg_kernel(const int* __restrict__ dst, float* __restrict__ deg, int E) {
  int e = blockIdx.x * blockDim.x + threadIdx.x;
  if (e < E) atomic_add_f32(&deg[dst[e]], 1.0f);
}

__global__ void invdeg_kernel(float* __restrict__ deg, int N) {
  int n = blockIdx.x * blockDim.x + threadIdx.x;
  if (n < N) deg[n] = 1.0f / fmaxf(deg[n], 1.0f);
}

// ------------------------------------------------------------- aggregate ---
// One wave (32 lanes) per edge; each lane handles 4 contiguous features.
__global__ void aggregate_kernel(const float* __restrict__ x,
                                 const float* __restrict__ ew,
                                 const int* __restrict__ src,
                                 const int* __restrict__ dst,
                                 float* __restrict__ agg, int E) {
  long long t = (long long)blockIdx.x * blockDim.x + threadIdx.x;
  int e    = (int)(t >> 5);
  int lane = (int)(t & 31);
  if (e >= E) return;
  int   s = src[e];
  int   d = dst[e];
  float w = ew[e];
  float4 v = ((const float4*)(x + (size_t)s * FEATS))[lane];
  float* out = agg + (size_t)d * FEATS + lane * 4;
  atomic_add_f32(out + 0, v.x * w);
  atomic_add_f32(out + 1, v.y * w);
  atomic_add_f32(out + 2, v.z * w);
  atomic_add_f32(out + 3, v.w * w);
}

// ------------------------------------------------------------------ GEMM ---
// Out[row0:row0+16, col0:col0+16] = X*Wself + (AGG*invdeg)*Wneigh + bias
// Block = 256 threads = 8 waves; wave w owns column tile w (8*16 = 128 cols).
// K-loop steps by 4 using V_WMMA_F32_16X16X4_F32 (fp32 in, fp32 acc).
//
// A fragment (16x4 f32): lane l<16 holds row M=l, K={k,k+1}; lane l>=16 holds
// row M=l-16, K={k+2,k+3}  -> one contiguous float2 load per lane.
// B fragment (4x16 f32): mirrored; column col0+(l&15), rows k+hi*2, k+1+hi*2.
// C/D: VGPR v, lane l<16 -> (M=v,     N=l);  lane l>=16 -> (M=v+8, N=l-16).
template <bool RELU>
__global__ void sage_gemm_kernel(const float* __restrict__ X,       // [N,128]
                                 const float* __restrict__ AGG,     // [N,128]
                                 const float* __restrict__ invdeg,  // [N]
                                 const float* __restrict__ Wself,   // [128,128]
                                 const float* __restrict__ Wneigh,  // [128,128]
                                 const float* __restrict__ bias,    // [128]
                                 float* __restrict__ Out) {         // [N,128]
  const int wave = threadIdx.x >> 5;
  const int lane = threadIdx.x & 31;
  const int hi   = lane >> 4;     // 0: K pair {k,k+1}, 1: {k+2,k+3}
  const int l15  = lane & 15;
  const int row0 = blockIdx.x * 16;
  const int col0 = wave * 16;

  const int   row = row0 + l15;
  const float idg = invdeg[row];

  const float* xrow  = X      + (size_t)row * FEATS + hi * 2;
  const float* arow  = AGG    + (size_t)row * FEATS + hi * 2;
  const float* wscol = Wself  + (size_t)(hi * 2) * FEATS + col0 + l15;
  const float* wncol = Wneigh + (size_t)(hi * 2) * FEATS + col0 + l15;

  v8f c = {};
#pragma unroll 8
  for (int k = 0; k < FEATS; k += 4) {
    v2f a_s = *(const v2f*)(xrow + k);       // X[row][k+hi*2 .. +1]
    v2f a_n = *(const v2f*)(arow + k) * idg; // mean-normalized neighbor feats

    v2f b_s, b_n;
    b_s.x = wscol[(size_t)k * FEATS];
    b_s.y = wscol[(size_t)(k + 1) * FEATS];
    b_n.x = wncol[(size_t)k * FEATS];
    b_n.y = wncol[(size_t)(k + 1) * FEATS];

    c = __builtin_amdgcn_wmma_f32_16x16x4_f32(false, a_s, false, b_s,
                                              (short)0, c, false, false);
    c = __builtin_amdgcn_wmma_f32_16x16x4_f32(false, a_n, false, b_n,
                                              (short)0, c, false, false);
  }

  const float bv = bias[col0 + l15];
#pragma unroll
  for (int v = 0; v < 8; ++v) {
    int   m = v + hi * 8;
    float o = c[v] + bv;
    if (RELU) o = fmaxf(o, 0.0f);
    Out[(size_t)(row0 + m) * FEATS + col0 + l15] = o;
  }
}

// ---------------------------------------------------------------- launch ---
extern "C" void kernel_launch(void* const* d_in, const int* in_sizes, int n_in,
                              void* d_out, int out_size, void* d_ws, size_t ws_size,
                              hipStream_t stream) {
  const float* in_feat = (const float*)d_in[0];
  const float* ew      = (const float*)d_in[1];
  const float* W1s     = (const float*)d_in[2];
  const float* W1n     = (const float*)d_in[3];
  const float* b1      = (const float*)d_in[4];
  const float* W2s     = (const float*)d_in[5];
  const float* W2n     = (const float*)d_in[6];
  const float* b2      = (const float*)d_in[7];
  const int*   src     = (const int*)d_in[8];
  const int*   dst     = (const int*)d_in[9];

  const int E = in_sizes[1];   // 1,600,000
  const int N = NNODES;        // 50,000 (divisible by 16)

  // Workspace layout: [deg/invdeg N][agg N*128][h1 N*128]  (~51.4 MB)
  char*  ws  = (char*)d_ws;
  float* deg = (float*)ws;
  size_t o1  = ((size_t)N * 4 + 255) & ~(size_t)255;
  float* agg = (float*)(ws + o1);
  size_t o2  = o1 + (((size_t)N * FEATS * 4 + 255) & ~(size_t)255);
  float* h1  = (float*)(ws + o2);

  const int TPB = 256;

  // degree / inverse degree
  hipMemsetAsync(deg, 0, (size_t)N * 4, stream);
  deg_kernel<<<(E + TPB - 1) / TPB, TPB, 0, stream>>>(dst, deg, E);
  invdeg_kernel<<<(N + TPB - 1) / TPB, TPB, 0, stream>>>(deg, N);

  const long long aggThreads = (long long)E * 32;
  const int aggBlocks = (int)((aggThreads + TPB - 1) / TPB);

  // ----- layer 1 -----
  hipMemsetAsync(agg, 0, (size_t)N * FEATS * 4, stream);
  aggregate_kernel<<<aggBlocks, TPB, 0, stream>>>(in_feat, ew, src, dst, agg, E);
  sage_gemm_kernel<true><<<N / 16, TPB, 0, stream>>>(in_feat, agg, deg,
                                                     W1s, W1n, b1, h1);

  // ----- layer 2 -----
  hipMemsetAsync(agg, 0, (size_t)N * FEATS * 4, stream);
  aggregate_kernel<<<aggBlocks, TPB, 0, stream>>>(h1, ew, src, dst, agg, E);
  sage_gemm_kernel<false><<<N / 16, TPB, 0, stream>>>(h1, agg, deg,
                                                      W2s, W2n, b2, (float*)d_out);
}